// ROIPool_18167711662403
// MI455X (gfx1250) — compile-verified
//
#include <hip/hip_runtime.h>

// ROI-Align as per-ROI GEMM on CDNA5 WMMA (V_WMMA_F32_16X16X4_F32, full f32 precision).
//
// out[r, c, oy, ox] = sum_{ty,tx} Wy[oy,ty] * Wx[ox,tx] * F[b, c, Y[ty], X[tx]]
// Wy/Wx fold bilinear weights, validity mask and the 1/4 pooling mean.
// M = 49 cells (padded to 64), N = 16 channels per workgroup, K = Ty*Tx <= 784 (typ ~225).

#define OUTD   7
#define SGRID  14          // OUT * RATIO
#define MAXT   28          // max unique taps per axis (14 samples x 2 taps)
#define KMAX   (MAXT*MAXT) // 784
#define CN     256
#define HH     200
#define WW     304
#define NROI   1024
#define CG     16          // channels per workgroup (WMMA N)
#define TPB    128         // 4 waves (wave32)

typedef float v2f __attribute__((ext_vector_type(2)));
typedef float v8f __attribute__((ext_vector_type(8)));

__global__ __launch_bounds__(TPB) void roi_align_wmma_f32(
    const float* __restrict__ feat,
    const float* __restrict__ boxes,
    float* __restrict__ out)
{
    __shared__ float        sV[KMAX * CG];      // gathered tile, k-major: sV[k*CG + c] (50176 B)
    __shared__ unsigned int sK[KMAX];           // k -> (ty | tx<<16)
    __shared__ float        sWy[OUTD * MAXT];   // y-axis weights (incl. 1/2 * valid)
    __shared__ float        sWx[OUTD * MAXT];   // x-axis weights (incl. 1/2 * valid)
    __shared__ int          sYT[MAXT];          // unique y taps (strictly increasing)
    __shared__ int          sXT[MAXT];          // unique x taps (strictly increasing)
    __shared__ int          sMeta[4];

    const int tid = threadIdx.x;
    const int r   = blockIdx.y;          // ROI index 0..1023
    const int c0  = blockIdx.x * CG;     // channel group base

    // ---- 1) cooperative zero of weight matrices ----
    for (int i = tid; i < OUTD * MAXT; i += TPB) { sWy[i] = 0.f; sWx[i] = 0.f; }
    __syncthreads();

    // ---- 2) serial per-ROI tap + weight build (torchvision roi_align semantics).
    // Sample coords are monotone in s, so the tap list grows strictly increasing and
    // dedup only needs to inspect the last two entries (compact code, no search loop).
    if (tid == 0) {
        const float* bx = boxes + (size_t)r * 4;
        float x1 = bx[0] * 0.25f, y1 = bx[1] * 0.25f;
        float x2 = bx[2] * 0.25f, y2 = bx[3] * 0.25f;
        float roi_w = fmaxf(x2 - x1, 1.0f);
        float roi_h = fmaxf(y2 - y1, 1.0f);
        float stepx = roi_w * (1.0f / (float)SGRID);
        float stepy = roi_h * (1.0f / (float)SGRID);

        int tyn = 0, txn = 0;
#pragma unroll 1
        for (int s = 0; s < SGRID; s++) {       // Y axis
            float t     = y1 + ((float)s + 0.5f) * stepy;
            bool  valid = (t > -1.0f) && (t < (float)HH);
            float tc    = (t < 0.f) ? 0.f : t;
            float low   = floorf(tc);
            bool  edge  = (low >= (float)(HH - 1));
            int   lo    = edge ? (HH - 1) : (int)low;
            int   hi    = edge ? (HH - 1) : ((int)low + 1);
            float fr    = edge ? 0.f : (tc - low);
            if (valid) {
                int il;
                if      (tyn >= 1 && sYT[tyn - 1] == lo) il = tyn - 1;
                else if (tyn >= 2 && sYT[tyn - 2] == lo) il = tyn - 2;
                else { il = tyn; sYT[tyn++] = lo; }
                int ih;
                if      (hi == lo)                       ih = il;
                else if (tyn >= 1 && sYT[tyn - 1] == hi) ih = tyn - 1;
                else { ih = tyn; sYT[tyn++] = hi; }
                int oy = s >> 1;
                sWy[oy * MAXT + il] += 0.5f * (1.0f - fr);
                sWy[oy * MAXT + ih] += 0.5f * fr;
            }
        }
#pragma unroll 1
        for (int s = 0; s < SGRID; s++) {       // X axis
            float t     = x1 + ((float)s + 0.5f) * stepx;
            bool  valid = (t > -1.0f) && (t < (float)WW);
            float tc    = (t < 0.f) ? 0.f : t;
            float low   = floorf(tc);
            bool  edge  = (low >= (float)(WW - 1));
            int   lo    = edge ? (WW - 1) : (int)low;
            int   hi    = edge ? (WW - 1) : ((int)low + 1);
            float fr    = edge ? 0.f : (tc - low);
            if (valid) {
                int il;
                if      (txn >= 1 && sXT[txn - 1] == lo) il = txn - 1;
                else if (txn >= 2 && sXT[txn - 2] == lo) il = txn - 2;
                else { il = txn; sXT[txn++] = lo; }
                int ih;
                if      (hi == lo)                       ih = il;
                else if (txn >= 1 && sXT[txn - 1] == hi) ih = txn - 1;
                else { ih = txn; sXT[txn++] = hi; }
                int ox = s >> 1;
                sWx[ox * MAXT + il] += 0.5f * (1.0f - fr);
                sWx[ox * MAXT + ih] += 0.5f * fr;
            }
        }
        if (tyn == 0) { sYT[0] = 0; tyn = 1; }  // all-invalid: weights stay 0 -> output 0
        if (txn == 0) { sXT[0] = 0; txn = 1; }
        sMeta[0] = tyn;
        sMeta[1] = txn;
        sMeta[2] = r >> 9;                       // batch index (512 boxes per batch)
    }
    __syncthreads();

    const int tyn  = sMeta[0];
    const int txn  = sMeta[1];
    const int b    = sMeta[2];
    const int Ktot = tyn * txn;

    // ---- 3a) k -> (ty,tx) packed table ----
    for (int k = tid; k < Ktot; k += TPB) {
        int ty = k / txn;
        int tx = k - ty * txn;
        sK[k] = (unsigned int)ty | ((unsigned int)tx << 16);
    }
    __syncthreads();

    // ---- 3b) gather V tile (each unique tap loaded exactly once per WG).
    // 8 threads per channel sweep consecutive k: lanes hit consecutive taps of the
    // same feature-row segment -> good L2 locality; no integer division.
    {
        const int c     = tid >> 3;          // 0..15
        const int kslot = tid & 7;           // 0..7
        const float* fplane = feat + ((size_t)b * CN + (size_t)(c0 + c)) * (HH * WW);
        for (int k = kslot; k < Ktot; k += 8) {
            unsigned int pk = sK[k];
            int y = sYT[pk & 0xFFFFu];
            int x = sXT[pk >> 16];
            sV[k * CG + c] = fplane[y * WW + x];
        }
    }
    __syncthreads();

    // ---- 4) WMMA GEMM: D[cell, c] += A[cell, k] * B[k, c], K steps of 4 (f32) ----
    const int wave = tid >> 5;            // M-tile index 0..3 (cells 16*wave .. +15)
    const int lane = tid & 31;
    const int half = lane >> 4;           // A/B VGPR layout: K = 2*half + vgpr
    const int ln   = lane & 15;           // A: M = ln ; B: N(channel) = ln
    const int Mrow = wave * 16 + ln;      // output cell for this lane's A rows
    const int   oy   = (Mrow < 49) ? (Mrow / 7) : 0;
    const int   ox   = (Mrow < 49) ? (Mrow % 7) : 0;
    const float mval = (Mrow < 49) ? 1.f : 0.f;
    const float* wyRow = sWy + oy * MAXT;
    const float* wxRow = sWx + ox * MAXT;

    const int Ksteps = (Ktot + 3) >> 2;
    v8f acc = {0.f, 0.f, 0.f, 0.f, 0.f, 0.f, 0.f, 0.f};

    for (int ks = 0; ks < Ksteps; ks++) {
        int kb = ks * 4 + 2 * half;
        v2f a, bm;
#pragma unroll
        for (int v = 0; v < 2; v++) {
            int k  = kb + v;
            int kc = (k < Ktot) ? k : (Ktot - 1);    // clamp: padded K reads valid LDS
            unsigned int pk = sK[kc];
            float aw = wyRow[pk & 0xFFFFu] * wxRow[pk >> 16] * mval;
            a[v]  = (k < Ktot) ? aw : 0.f;           // zero-weight the K padding
            bm[v] = sV[kc * CG + ln];                // direct k-major B read
        }
        // D = A(16x4 f32) * B(4x16 f32) + C(16x16 f32)
        acc = __builtin_amdgcn_wmma_f32_16x16x4_f32(
                  false, a, false, bm, (short)0, acc, false, false);
    }

    // ---- 5) stage D through LDS, coalesced store to (R, C, 7, 7) ----
    __syncthreads();                       // everyone done reading sV
    float* sD = sV;                        // reuse tile LDS: sD[cell][c]
#pragma unroll
    for (int v = 0; v < 8; v++) {
        int cell = wave * 16 + v + 8 * half;   // C/D layout: vgpr v -> M = v + 8*(lane/16)
        sD[cell * CG + ln] = acc[v];
    }
    __syncthreads();

    const size_t obase = ((size_t)r * CN + c0) * (OUTD * OUTD);
    for (int i = tid; i < CG * (OUTD * OUTD); i += TPB) {
        int c    = i / 49;
        int cell = i - c * 49;             // consecutive threads -> consecutive cells
        out[obase + (size_t)c * 49 + cell] = sD[cell * CG + c];
    }
}

extern "C" void kernel_launch(void* const* d_in, const int* in_sizes, int n_in,
                              void* d_out, int out_size, void* d_ws, size_t ws_size,
                              hipStream_t stream) {
    const float* feat  = (const float*)d_in[0];   // (2, 256, 200, 304) f32
    const float* boxes = (const float*)d_in[1];   // (2, 512, 4) f32
    float* out = (float*)d_out;                   // (1024, 256, 7, 7) f32

    dim3 grid(CN / CG, NROI);                     // 16 channel-groups x 1024 ROIs
    roi_align_wmma_f32<<<grid, TPB, 0, stream>>>(feat, boxes, out);
}